// decode_87247965651294
// MI455X (gfx1250) — compile-verified
//
#include <hip/hip_runtime.h>
#include <stdint.h>

// Problem constants (match reference: cls_pred [32,128,128,80], loc_pred [32,128,128,4])
#define NB    32
#define HH    128
#define WW    128
#define CC    80
#define HWSZ  (HH * WW)          // 16384
#define NPB   (HWSZ * CC)        // 1,310,720 scores per batch
#define KTOP  100

#define BINS      4096           // top-12 bits of monotonic key
#define BIN_SHIFT 20
#define CAP       2048           // candidate capacity per batch
#define SORTN     2048           // bitonic sort size (== CAP)
#define TPB       256
#define BX        40             // streaming blocks per batch
#define ITERS     (NPB / (BX * TPB * 4))   // 32 float4-iterations per block

// ---- order-preserving float <-> uint key ----------------------------------
__device__ __forceinline__ uint32_t fkey(float f) {
    uint32_t x = __float_as_uint(f);
    // negative: flip all bits; positive: flip sign bit -> unsigned order == float order
    return x ^ (uint32_t)(((int32_t)x >> 31) | 0x80000000);
}
__device__ __forceinline__ float keyf(uint32_t u) {
    uint32_t x = (u & 0x80000000u) ? (u ^ 0x80000000u) : ~u;
    return __uint_as_float(x);
}

// ---- kernel 0: zero histogram + per-batch candidate counters ---------------
__global__ void k_init(uint32_t* __restrict__ hist, uint32_t* __restrict__ cnt) {
    int stride = gridDim.x * blockDim.x;
    for (int i = blockIdx.x * blockDim.x + threadIdx.x; i < NB * BINS; i += stride)
        hist[i] = 0u;
    if (blockIdx.x == 0 && threadIdx.x < NB)
        cnt[threadIdx.x] = 0u;
}

// ---- kernel 1: streaming histogram of key top-12 bits ----------------------
// CDNA5 path: double-buffered global_load_async_to_lds_b128 staging (ASYNCcnt),
// s_wait_asynccnt 0x1 overlaps the in-flight tile's HBM latency with key
// computation + DS-atomic LDS histogram of the landed tile. Per-lane staging
// slots => per-wave waits suffice (no block barrier in the hot loop).
__global__ void __launch_bounds__(TPB) k_hist(const float* __restrict__ cls,
                                              uint32_t* __restrict__ hist) {
    __shared__ float    stage[2][TPB * 4]; // 2 x 4 KB staging buffers
    __shared__ uint32_t lhist[BINS];       // 16 KB LDS histogram

    const int tid = threadIdx.x;
    const int b   = blockIdx.y;

    for (int i = tid; i < BINS; i += TPB) lhist[i] = 0u;
    __syncthreads();

    const float* sbase = cls + (size_t)b * NPB;            // uniform -> SGPR pair
    const uint32_t slot = (uint32_t)(blockIdx.x * TPB + tid); // float4 slot index
    // LDS byte offsets of this lane's two staging slots (flat LDS addr low 32 bits)
    uint32_t ldsAddr0 = (uint32_t)(size_t)(&stage[0][tid * 4]);
    uint32_t ldsAddr1 = (uint32_t)(size_t)(&stage[1][tid * 4]);

    // prologue: issue tile 0
    {
        uint32_t byteOff = slot * 16u;
        asm volatile("global_load_async_to_lds_b128 %0, %1, %2"
                     :: "v"(ldsAddr0), "v"(byteOff), "s"(sbase) : "memory");
    }

    for (int it = 0; it < ITERS; ++it) {
        if (it + 1 < ITERS) {
            // the buffer we are about to overwrite was last read at it-1:
            // make the WAR airtight, then issue the next tile and wait for
            // the *previous* one only (in-order completion => <=1 outstanding
            // means tile `it` has landed in LDS).
            asm volatile("s_wait_dscnt 0x0" ::: "memory");
            uint32_t byteOff = (slot + (uint32_t)(it + 1) * (BX * TPB)) * 16u;
            uint32_t dstAddr = ((it + 1) & 1) ? ldsAddr1 : ldsAddr0;
            asm volatile("global_load_async_to_lds_b128 %0, %1, %2"
                         :: "v"(dstAddr), "v"(byteOff), "s"(sbase) : "memory");
            asm volatile("s_wait_asynccnt 0x1" ::: "memory");
        } else {
            asm volatile("s_wait_asynccnt 0x0" ::: "memory");
        }
        const float4 v = *reinterpret_cast<const float4*>(&stage[it & 1][tid * 4]);
        atomicAdd(&lhist[fkey(v.x) >> BIN_SHIFT], 1u);
        atomicAdd(&lhist[fkey(v.y) >> BIN_SHIFT], 1u);
        atomicAdd(&lhist[fkey(v.z) >> BIN_SHIFT], 1u);
        atomicAdd(&lhist[fkey(v.w) >> BIN_SHIFT], 1u);
    }
    __syncthreads();

    for (int i = tid; i < BINS; i += TPB) {
        uint32_t v = lhist[i];
        if (v) atomicAdd(&hist[(size_t)b * BINS + i], v);
    }
}

// ---- kernel 2: find per-batch threshold bin P ------------------------------
// bins > P hold < KTOP elements  =>  bins >= P are a superset of the top-K.
__global__ void k_scan(const uint32_t* __restrict__ hist, uint32_t* __restrict__ Pout) {
    __shared__ uint32_t seg[TPB];
    const int b = blockIdx.x, tid = threadIdx.x;
    const int SEG = BINS / TPB;           // 16 bins per thread

    uint32_t s = 0;
    for (int i = 0; i < SEG; ++i)
        s += hist[(size_t)b * BINS + tid * SEG + i];
    seg[tid] = s;
    __syncthreads();

    if (tid == 0) {
        uint32_t cum = 0;
        int sgi = TPB - 1;
        for (; sgi > 0; --sgi) {
            if (cum + seg[sgi] >= KTOP) break;
            cum += seg[sgi];
        }
        int lo = sgi * SEG, hi = lo + SEG - 1;
        int P = lo;
        for (int bin = hi; bin >= lo; --bin) {
            uint32_t h = hist[(size_t)b * BINS + bin];
            if (cum + h >= KTOP) { P = bin; break; }
            cum += h;
        }
        Pout[b] = (uint32_t)P;
    }
}

// ---- kernel 3: compact candidates (key>=bin P) -----------------------------
// Second streaming read of cls_pred: mostly L2 hits (167.8 MB fits in 192 MB).
__global__ void __launch_bounds__(TPB) k_compact(const float* __restrict__ cls,
                                                 const uint32_t* __restrict__ Pin,
                                                 uint32_t* __restrict__ cnt,
                                                 unsigned long long* __restrict__ cand) {
    const int tid = threadIdx.x;
    const int b   = blockIdx.y;
    const uint32_t P = Pin[b];
    const float* sbase = cls + (size_t)b * NPB;
    const uint32_t slot = (uint32_t)(blockIdx.x * TPB + tid);

    for (int it = 0; it < ITERS; ++it) {
        uint32_t e0 = (slot + (uint32_t)it * (BX * TPB)) * 4u;   // element index of v.x
        const float4 v = *reinterpret_cast<const float4*>(sbase + e0);
        if (it + 1 < ITERS)
            __builtin_prefetch(sbase + e0 + (uint32_t)(BX * TPB) * 4u, 0, 1); // global_prefetch_b8
        float fv[4] = { v.x, v.y, v.z, v.w };
#pragma unroll
        for (int e = 0; e < 4; ++e) {
            uint32_t u = fkey(fv[e]);
            if ((u >> BIN_SHIFT) >= P) {
                uint32_t pos = atomicAdd(&cnt[b], 1u);
                if (pos < CAP) {
                    uint32_t idx = e0 + (uint32_t)e;
                    // composite key: score desc, then index asc (via ~idx)
                    cand[(size_t)b * CAP + pos] =
                        ((unsigned long long)u << 32) |
                        (unsigned long long)(0xFFFFFFFFu - idx);
                }
            }
        }
    }
}

// ---- kernel 4: per-batch bitonic sort + decode + gather --------------------
__global__ void __launch_bounds__(TPB) k_final(const float* __restrict__ locp,
                                               const uint32_t* __restrict__ cnt,
                                               const unsigned long long* __restrict__ cand,
                                               float* __restrict__ out) {
    __shared__ unsigned long long skey[SORTN];   // 16 KB
    const int tid = threadIdx.x;
    const int b   = blockIdx.x;

    uint32_t n = cnt[b];
    if (n > CAP) n = CAP;
    for (int i = tid; i < SORTN; i += TPB)
        skey[i] = (i < (int)n) ? cand[(size_t)b * CAP + i] : 0ull;
    __syncthreads();

    // bitonic sort, descending
    for (unsigned k = 2; k <= SORTN; k <<= 1) {
        for (unsigned j = k >> 1; j > 0; j >>= 1) {
            for (unsigned i = tid; i < SORTN; i += TPB) {
                unsigned ixj = i ^ j;
                if (ixj > i) {
                    unsigned long long a = skey[i], c = skey[ixj];
                    bool desc = ((i & k) == 0);
                    if (desc ? (a < c) : (a > c)) { skey[i] = c; skey[ixj] = a; }
                }
            }
            __syncthreads();
        }
    }

    if (tid < KTOP) {
        unsigned long long kk = skey[tid];
        uint32_t u   = (uint32_t)(kk >> 32);
        uint32_t idx = 0xFFFFFFFFu - (uint32_t)(kk & 0xFFFFFFFFu);
        float score  = keyf(u);
        uint32_t cid = idx % CC;
        uint32_t sp  = idx / CC;          // == ys*W + xs
        const float4 L = *reinterpret_cast<const float4*>(locp + ((size_t)b * HWSZ + sp) * 4);
        float* o = out + ((size_t)b * KTOP + (uint32_t)tid) * 6;
        o[0] = L.x; o[1] = L.y; o[2] = L.z; o[3] = L.w;   // SCALE == 1.0
        o[4] = score;
        o[5] = (float)cid;
    }
}

// ---- launcher --------------------------------------------------------------
extern "C" void kernel_launch(void* const* d_in, const int* in_sizes, int n_in,
                              void* d_out, int out_size, void* d_ws, size_t ws_size,
                              hipStream_t stream) {
    (void)in_sizes; (void)n_in; (void)out_size; (void)ws_size;
    const float* cls = (const float*)d_in[0];   // [32,128,128,80] f32
    const float* loc = (const float*)d_in[1];   // [32,128,128,4]  f32
    float* out = (float*)d_out;                 // [32,100,6]      f32

    uint8_t* ws = (uint8_t*)d_ws;
    uint32_t*           hist = (uint32_t*)(ws);                      // 512 KB
    uint32_t*           P    = (uint32_t*)(ws + 524288);             // 128 B
    uint32_t*           cnt  = (uint32_t*)(ws + 524288 + 256);       // 128 B
    unsigned long long* cand = (unsigned long long*)(ws + 524288 + 512); // 512 KB

    k_init   <<<dim3(128),      dim3(TPB), 0, stream>>>(hist, cnt);
    k_hist   <<<dim3(BX, NB),   dim3(TPB), 0, stream>>>(cls, hist);
    k_scan   <<<dim3(NB),       dim3(TPB), 0, stream>>>(hist, P);
    k_compact<<<dim3(BX, NB),   dim3(TPB), 0, stream>>>(cls, P, cnt, cand);
    k_final  <<<dim3(NB),       dim3(TPB), 0, stream>>>(loc, cnt, cand, out);
}